// BatchSparseMultiHeadGraphAttention_6828998001375
// MI455X (gfx1250) — compile-verified
//
#include <hip/hip_runtime.h>
#include <math.h>

typedef float v2f __attribute__((ext_vector_type(2)));
typedef float v8f __attribute__((ext_vector_type(8)));
typedef unsigned int u32x4 __attribute__((ext_vector_type(4)));
typedef int i32x4 __attribute__((ext_vector_type(4)));
typedef int i32x8 __attribute__((ext_vector_type(8)));

constexpr int       Bc = 4;
constexpr int       Nc = 20000;
constexpr int       Fc = 256;     // F_IN == H*D
constexpr int       Hc = 8;
constexpr int       Dc = 32;
constexpr long long Ec = 320000;

#if defined(__has_builtin)
#if __has_builtin(__builtin_amdgcn_tensor_load_to_lds)
#define HAVE_TDM 1
#endif
#endif
#ifndef HAVE_TDM
#define HAVE_TDM 0
#endif

// ---- ordered-uint encoding for float atomicMax ----
__device__ __forceinline__ unsigned enc_f32(float f) {
    unsigned u = __float_as_uint(f);
    return (u & 0x80000000u) ? ~u : (u | 0x80000000u);
}
__device__ __forceinline__ float dec_f32(unsigned e) {
    unsigned u = (e & 0x80000000u) ? (e & 0x7FFFFFFFu) : ~e;
    return __uint_as_float(u);
}
__device__ __forceinline__ float lrelu(float x) { return x > 0.f ? x : 0.2f * x; }

// ---------------------------------------------------------------------------
// K0: zero output, denom, batch-max
// ---------------------------------------------------------------------------
__global__ void k_zero(float* __restrict__ out, float* __restrict__ denom,
                       unsigned* __restrict__ bmax) {
    long long i = (long long)blockIdx.x * blockDim.x + threadIdx.x;
    if (i < (long long)Bc * Hc * Nc * Dc) out[i] = 0.f;
    if (i < (long long)Bc * Nc * Hc)      denom[i] = 0.f;
    if (i < Bc)                           bmax[i] = 0u;   // enc identity for max
}

// ---------------------------------------------------------------------------
// K1: h_prime = h @ W^T  via V_WMMA_F32_16X16X4_F32
//   block = 256 thr (8 waves); block owns one (batch, 16-row strip)
//   wave w owns head w (two 16-col tiles).
//   A strip staged to LDS by the Tensor Data Mover (2D D#, HW padding gives
//   the 260-float conflict-free row stride); TENSORcnt-gated.
// ---------------------------------------------------------------------------
__global__ __launch_bounds__(256)
void k_gemm(const float* __restrict__ h, const float* __restrict__ W,
            float* __restrict__ hp) {
    constexpr int STRIDE = 260;           // 256 data + 4 pad dwords per row
    __shared__ float sA[16 * STRIDE];

    const int strip = blockIdx.x % (Nc / 16);
    const int b     = blockIdx.x / (Nc / 16);
    const int n0    = strip * 16;
    const int t     = threadIdx.x;

    const float* hb = h + ((size_t)b * Nc + n0) * Fc;

#if HAVE_TDM
    if (t < 32) {   // wave 0 issues one TDM descriptor for the 16x256 tile
        unsigned            ldsb = (unsigned)(size_t)&sA[0];      // addr[31:0] == LDS byte addr
        unsigned long long  ga   = (unsigned long long)(const void*)hb;
        u32x4 g0;
        g0[0] = 1u;                                               // count=1, user mode
        g0[1] = ldsb;                                             // lds_addr
        g0[2] = (unsigned)(ga & 0xFFFFFFFFull);                   // global_addr[31:0]
        g0[3] = (unsigned)((ga >> 32) & 0x1FFFFFFull) | (2u << 30); // addr[56:32] | type=2
        i32x8 g1;
        g1[0] = (int)((2u << 16)        // data_size = 4B
                    | (1u << 20)        // pad_enable
                    | (7u << 22)        // pad_interval = 256 DWORDs (one row)
                    | (3u << 25));      // pad_amount   = 4 DWORDs  -> stride 260
        g1[1] = (int)(256u << 16);      // tensor_dim0 = 256 (bits 79:48)
        g1[2] = (int)(16u << 16);       // tensor_dim1 = 16  (bits 111:80)
        g1[3] = (int)(256u << 16);      // tile_dim0   = 256 (bits 127:112)
        g1[4] = 16;                     // tile_dim1 = 16, tile_dim2 = 0
        g1[5] = 256;                    // tensor_dim0_stride = 256 elements
        g1[6] = 0;
        g1[7] = 0;
        i32x4 g2 = {0, 0, 0, 0};
        i32x4 g3 = {0, 0, 0, 0};
#if defined(__clang_major__) && (__clang_major__ >= 23)
        i32x8 g4 = {0, 0, 0, 0, 0, 0, 0, 0};
        __builtin_amdgcn_tensor_load_to_lds(g0, g1, g2, g3, g4, 0);
#else
        __builtin_amdgcn_tensor_load_to_lds(g0, g1, g2, g3, 0);
#endif
        __builtin_amdgcn_s_wait_tensorcnt(0);
    }
#else
    // cooperative 16x256 load (1024 float4s / 256 threads)
#pragma unroll
    for (int i = 0; i < 4; ++i) {
        int idx = t + i * 256;
        int row = idx >> 6;
        int c4  = (idx & 63) << 2;
        float4 v = *(const float4*)(hb + row * Fc + c4);
        float* dst = &sA[row * STRIDE + c4];
        dst[0] = v.x; dst[1] = v.y; dst[2] = v.z; dst[3] = v.w;
    }
#endif
    __syncthreads();

    const int lane = t & 31;
    const int wave = t >> 5;          // == head index
    const int half = lane >> 4;       // K sub-pair select (0:{0,1}, 1:{2,3})
    const int l15  = lane & 15;

    const float* arow  = &sA[l15 * STRIDE + 2 * half];
    const float* wrow0 = W + (((wave * 2 + 0) * 16 + l15) * Fc) + 2 * half;
    const float* wrow1 = W + (((wave * 2 + 1) * 16 + l15) * Fc) + 2 * half;
    __builtin_prefetch(wrow0, 0, 0);
    __builtin_prefetch(wrow1, 0, 0);

    v8f acc0 = {};
    v8f acc1 = {};
#pragma unroll 8
    for (int k0 = 0; k0 < Fc; k0 += 4) {
        v2f a  = *(const v2f*)(arow  + k0);
        v2f b0 = *(const v2f*)(wrow0 + k0);
        v2f b1 = *(const v2f*)(wrow1 + k0);
        acc0 = __builtin_amdgcn_wmma_f32_16x16x4_f32(false, a, false, b0,
                                                     (short)0, acc0, false, false);
        acc1 = __builtin_amdgcn_wmma_f32_16x16x4_f32(false, a, false, b1,
                                                     (short)0, acc1, false, false);
    }

    // C/D layout: VGPR i -> row n0 + i + 8*half, col = tile_col*16 + l15
    float* out0 = hp + ((size_t)b * Nc + n0 + 8 * half) * Fc + (wave * 32 + l15);
#pragma unroll
    for (int i = 0; i < 8; ++i) {
        out0[(size_t)i * Fc]      = acc0[i];
        out0[(size_t)i * Fc + 16] = acc1[i];
    }
}

// ---------------------------------------------------------------------------
// K2: a_src[b,n,h] = <h'[b,n,h,:], attn_src[h,:]>, same for a_trg
// ---------------------------------------------------------------------------
__global__ void k_proj(const float* __restrict__ hp, const float* __restrict__ w_s,
                       const float* __restrict__ w_t, float* __restrict__ asrc,
                       float* __restrict__ atrg) {
    long long i = (long long)blockIdx.x * blockDim.x + threadIdx.x; // (b*N+n)*H+h
    if (i >= (long long)Bc * Nc * Hc) return;
    int hh = (int)(i & 7);
    long long bn = i >> 3;
    const float4* p4 = (const float4*)(hp + bn * Fc + hh * Dc);
    const float4* s4 = (const float4*)(w_s + hh * Dc);
    const float4* t4 = (const float4*)(w_t + hh * Dc);
    float s = 0.f, tt = 0.f;
#pragma unroll
    for (int d = 0; d < Dc / 4; ++d) {
        float4 v = p4[d], a = s4[d], c = t4[d];
        s  += v.x * a.x + v.y * a.y + v.z * a.z + v.w * a.w;
        tt += v.x * c.x + v.y * c.y + v.z * c.z + v.w * c.w;
    }
    asrc[i] = s;
    atrg[i] = tt;
}

// ---------------------------------------------------------------------------
// K3: per-batch global max of leaky_relu(a_src[src]+a_trg[trg]) over [E,H]
// ---------------------------------------------------------------------------
__global__ __launch_bounds__(256)
void k_max(const float* __restrict__ asrc, const float* __restrict__ atrg,
           const int* __restrict__ ei, unsigned* __restrict__ bmax) {
    __shared__ unsigned red[256];
    const int blocksPerBatch = (int)(Ec / 256);
    const int b = blockIdx.x / blocksPerBatch;
    long long e = (long long)(blockIdx.x % blocksPerBatch) * 256 + threadIdx.x;

    int src = ei[(long long)b * 2 * Ec + e];
    int trg = ei[(long long)b * 2 * Ec + Ec + e];
    const float* ps = asrc + ((long long)b * Nc + src) * Hc;
    const float* pt = atrg + ((long long)b * Nc + trg) * Hc;
    float m = -INFINITY;
#pragma unroll
    for (int hh = 0; hh < Hc; ++hh) m = fmaxf(m, lrelu(ps[hh] + pt[hh]));

    red[threadIdx.x] = enc_f32(m);
    __syncthreads();
    for (int s = 128; s > 0; s >>= 1) {
        if (threadIdx.x < (unsigned)s)
            red[threadIdx.x] = max(red[threadIdx.x], red[threadIdx.x + s]);
        __syncthreads();
    }
    if (threadIdx.x == 0) atomicMax(&bmax[b], red[0]);
}

// ---------------------------------------------------------------------------
// K4: denom[b,trg,h] += exp(e - max_b)
// ---------------------------------------------------------------------------
__global__ void k_denom(const float* __restrict__ asrc, const float* __restrict__ atrg,
                        const int* __restrict__ ei, const unsigned* __restrict__ bmax,
                        float* __restrict__ denom) {
    long long i = (long long)blockIdx.x * blockDim.x + threadIdx.x; // (b*E+e)*H+h
    if (i >= (long long)Bc * Ec * Hc) return;
    int hh = (int)(i & 7);
    long long be = i >> 3;
    int b = (int)(be / Ec);
    long long e = be - (long long)b * Ec;
    int src = ei[(long long)b * 2 * Ec + e];
    int trg = ei[(long long)b * 2 * Ec + Ec + e];
    float x = lrelu(asrc[((long long)b * Nc + src) * Hc + hh] +
                    atrg[((long long)b * Nc + trg) * Hc + hh]);
    float ex = __expf(x - dec_f32(bmax[b]));
    atomicAdd(&denom[((long long)b * Nc + trg) * Hc + hh], ex);
}

// ---------------------------------------------------------------------------
// K5: out[b,h,trg,d] += h'[b,src,h,d] * attn ; one block per edge, t=(h,d)
// ---------------------------------------------------------------------------
__global__ __launch_bounds__(256)
void k_scatter(const float* __restrict__ hp, const float* __restrict__ asrc,
               const float* __restrict__ atrg, const int* __restrict__ ei,
               const unsigned* __restrict__ bmax, const float* __restrict__ denom,
               float* __restrict__ out) {
    long long be = blockIdx.x;                // b*E + e
    int b = (int)(be / Ec);
    long long e = be - (long long)b * Ec;
    int t  = threadIdx.x;
    int hh = t >> 5;
    int d  = t & 31;

    int src = ei[(long long)b * 2 * Ec + e];
    int trg = ei[(long long)b * 2 * Ec + Ec + e];
    if (t == 0) __builtin_prefetch(hp + ((long long)b * Nc + src) * Fc, 0, 0);

    float x = lrelu(asrc[((long long)b * Nc + src) * Hc + hh] +
                    atrg[((long long)b * Nc + trg) * Hc + hh]);
    float ex   = __expf(x - dec_f32(bmax[b]));
    float attn = ex / (denom[((long long)b * Nc + trg) * Hc + hh] + 1e-16f);
    float v = hp[(((long long)b * Nc + src) * Hc + hh) * Dc + d] * attn;
    atomicAdd(out + (((long long)b * Hc + hh) * Nc + trg) * Dc + d, v);
}

// ---------------------------------------------------------------------------
extern "C" void kernel_launch(void* const* d_in, const int* in_sizes, int n_in,
                              void* d_out, int out_size, void* d_ws, size_t ws_size,
                              hipStream_t stream) {
    (void)in_sizes; (void)n_in; (void)out_size; (void)ws_size;
    const float* h   = (const float*)d_in[0];
    const int*   ei  = (const int*)d_in[1];
    const float* W   = (const float*)d_in[2];
    const float* w_s = (const float*)d_in[3];
    const float* w_t = (const float*)d_in[4];
    float* out = (float*)d_out;

    float* hp    = (float*)d_ws;                         // B*N*F
    float* asrc  = hp   + (long long)Bc * Nc * Fc;       // B*N*H
    float* atrg  = asrc + (long long)Bc * Nc * Hc;       // B*N*H
    float* denom = atrg + (long long)Bc * Nc * Hc;       // B*N*H
    unsigned* bmax = (unsigned*)(denom + (long long)Bc * Nc * Hc); // B

    long long outElems = (long long)Bc * Hc * Nc * Dc;
    k_zero<<<(int)((outElems + 255) / 256), 256, 0, stream>>>(out, denom, bmax);
    k_gemm<<<Bc * (Nc / 16), 256, 0, stream>>>(h, W, hp);
    k_proj<<<(int)(((long long)Bc * Nc * Hc + 255) / 256), 256, 0, stream>>>(hp, w_s, w_t, asrc, atrg);
    k_max<<<(int)(Bc * (Ec / 256)), 256, 0, stream>>>(asrc, atrg, ei, bmax);
    k_denom<<<(int)(((long long)Bc * Ec * Hc + 255) / 256), 256, 0, stream>>>(asrc, atrg, ei, bmax, denom);
    k_scatter<<<(int)(Bc * Ec), 256, 0, stream>>>(hp, asrc, atrg, ei, bmax, denom, out);
}